// mGaussNorm_S_66915590472012
// MI455X (gfx1250) — compile-verified
//
#include <hip/hip_runtime.h>

typedef __attribute__((ext_vector_type(2))) float v2f;
typedef __attribute__((ext_vector_type(8))) float v8f;

#define CIN 64
#define H_SZ 4096
#define FD 8
#define HALF_T 36
#define LOC_ELEMS (32 * 4096 * 8)   // loc floats, pd follows
#define YPITCH 50                   // padded row (floats) to reduce LDS bank conflicts
#define NFRAG (128 * 3 * 64)        // 24576 floats: [kk][mt][64] A-fragment table

__device__ __forceinline__ float softplus_f(float x) {
  return fmaxf(x, 0.0f) + log1pf(expf(-fabsf(x)));
}

__global__ __launch_bounds__(256) void fused_gauss_kernel(
    const float* __restrict__ x, const float* __restrict__ w_n,
    const float* __restrict__ b_n, const float* __restrict__ w_p,
    const float* __restrict__ b_p, float* __restrict__ out)
{
  // 96 KB, used twice: phase 1 = WMMA A-fragment table, phase 2 = Y exchange
  __shared__ float smem[NFRAG];

  const int tid = threadIdx.x;

  // ---------------- build padded im2col weight matrix as lane-ordered A-fragments
  // entry e = kk*192 + mt*64 + (lane*2 + j):
  //   M (out channel) = mt*16 + (lane&15),  K (reduction r) = 4*kk + 2*(lane>>4) + j
  for (int e = tid; e < NFRAG; e += 256) {
    const int kk  = e / 192;
    const int rem = e - kk * 192;
    const int mt  = rem >> 6;
    const int f   = rem & 63;
    const int ll  = f >> 1, j = f & 1;
    const int outc = mt * 16 + (ll & 15);
    const int r    = 4 * kk + 2 * (ll >> 4) + j;
    const int c    = r >> 3, u = r & 7;
    float val = 0.0f;
    if (outc < HALF_T) {                       // p-conv channels (o, wo), k in [0,5)
      const int o = outc >> 2, wo = outc & 3, k = u - wo;
      if (k >= 0 && k < 5) val = w_p[(o * CIN + c) * 5 + k];
    } else if (outc < 44) {                    // loc channels: identity window * w_n
      if (u == outc - HALF_T) val = w_n[c];
    }
    smem[e] = val;
  }
  __syncthreads();

  const int lane = tid & 31;
  const int wv   = tid >> 5;     // wave 0..7
  const int n0   = lane & 15;
  const int kh   = lane >> 4;

  const int wg    = blockIdx.x;       // 512 blocks
  const int b     = wg >> 4;
  const int hbase = (wg & 15) << 8;
  const int h0    = hbase + wv * 32;  // this wave's 32 positions (2 WMMA tiles)

  const float* xb   = x + (size_t)b * CIN * (H_SZ * FD);
  const float* pxb  = xb + (size_t)(h0 + n0) * FD;
  const float* frag = &smem[lane * 2];

  // ---------------- GEMM: Y[48 outs x 32 positions] via v_wmma_f32_16x16x4_f32
  v8f acc[2][3] = {};
  const int ubase = kh << 1;

  #pragma unroll 4
  for (int kk = 0; kk < 128; ++kk) {
    const int c = kk >> 1;
    const int u = ((kk & 1) << 2) + ubase;
    // B fragments: x (K=4 x N=16 positions), one b64 per lane, streamed (NT)
    const float* px = pxb + (size_t)c * (H_SZ * FD) + u;
    v2f bt0 = __builtin_nontemporal_load((const v2f*)px);
    v2f bt1 = __builtin_nontemporal_load((const v2f*)(px + 16 * FD));
    // A fragments: 3 unconditional ds_load_b64 from the prebuilt table
    const float* fr = frag + kk * 192;
    v2f a0 = *(const v2f*)(fr);
    v2f a1 = *(const v2f*)(fr + 64);
    v2f a2 = *(const v2f*)(fr + 128);
    acc[0][0] = __builtin_amdgcn_wmma_f32_16x16x4_f32(false, a0, false, bt0, (short)0, acc[0][0], false, false);
    acc[1][0] = __builtin_amdgcn_wmma_f32_16x16x4_f32(false, a0, false, bt1, (short)0, acc[1][0], false, false);
    acc[0][1] = __builtin_amdgcn_wmma_f32_16x16x4_f32(false, a1, false, bt0, (short)0, acc[0][1], false, false);
    acc[1][1] = __builtin_amdgcn_wmma_f32_16x16x4_f32(false, a1, false, bt1, (short)0, acc[1][1], false, false);
    acc[0][2] = __builtin_amdgcn_wmma_f32_16x16x4_f32(false, a2, false, bt0, (short)0, acc[0][2], false, false);
    acc[1][2] = __builtin_amdgcn_wmma_f32_16x16x4_f32(false, a2, false, bt1, (short)0, acc[1][2], false, false);
  }

  // ---------------- redistribute D fragments -> one position per lane (reuse smem)
  __syncthreads();                    // everyone done reading the fragment table
  #pragma unroll
  for (int t0 = 0; t0 < 2; ++t0) {
    const int pos = wv * 32 + t0 * 16 + n0;
    #pragma unroll
    for (int mt = 0; mt < 3; ++mt) {
      float* dst = &smem[pos * YPITCH + mt * 16 + kh * 8];
      v8f v = acc[t0][mt];
      #pragma unroll
      for (int r = 0; r < 8; r += 2)
        *(float2*)(dst + r) = make_float2(v[r], v[r + 1]);
    }
  }
  __syncthreads();

  // ---------------- per-lane: softplus, symmetric 8x8, Jacobi, clamp, rebuild
  const int pos = wv * 32 + lane;
  const size_t gpos = (size_t)wg * 256 + pos;   // == b*4096 + h
  const float* yb = &smem[pos * YPITCH];

  float bp[9];
  #pragma unroll
  for (int o = 0; o < 9; ++o) bp[o] = b_p[o];
  const float bn = b_n[0];

  { // loc = softplus(1x1-conv + b_n)
    float l[8];
    #pragma unroll
    for (int w = 0; w < 8; ++w) l[w] = softplus_f(yb[HALF_T + w] + bn);
    float* lp = out + gpos * 8;
    *(float4*)lp       = make_float4(l[0], l[1], l[2], l[3]);
    *(float4*)(lp + 4) = make_float4(l[4], l[5], l[6], l[7]);
  }

  float A[8][8];
  { // scatter 36 softplus values symmetrically (row-major upper triangle)
    int t = 0;
    #pragma unroll
    for (int i = 0; i < 8; ++i)
      #pragma unroll
      for (int j = i; j < 8; ++j) {
        const float v = softplus_f(yb[t] + bp[t >> 2]);
        A[i][j] = v; A[j][i] = v;
        ++t;
      }
  }

  float V[8][8];
  #pragma unroll
  for (int i = 0; i < 8; ++i)
    #pragma unroll
    for (int j = 0; j < 8; ++j)
      V[i][j] = (i == j) ? 1.0f : 0.0f;

  // cyclic Jacobi, fully register-resident, branchless rotations
  #pragma unroll 1
  for (int sweep = 0; sweep < 8; ++sweep) {
    #pragma unroll
    for (int p = 0; p < 7; ++p) {
      #pragma unroll
      for (int q = p + 1; q < 8; ++q) {
        const float apq  = A[p][q];
        const bool  tiny = fabsf(apq) < 1e-28f;
        const float apqs = tiny ? 1.0f : apq;
        const float theta = 0.5f * (A[q][q] - A[p][p]) / apqs;
        float tt = copysignf(1.0f, theta) /
                   (fabsf(theta) + sqrtf(fmaf(theta, theta, 1.0f)));
        if (tiny) tt = 0.0f;
        const float cr = rsqrtf(fmaf(tt, tt, 1.0f));
        const float sr = tt * cr;
        const float app = A[p][p], aqq = A[q][q];
        A[p][p] = fmaf(-tt, apq, app);
        A[q][q] = fmaf( tt, apq, aqq);
        A[p][q] = 0.0f; A[q][p] = 0.0f;
        #pragma unroll
        for (int i = 0; i < 8; ++i) {
          if (i == p || i == q) continue;
          const float aip = A[i][p], aiq = A[i][q];
          const float nip = fmaf(cr, aip, -sr * aiq);
          const float niq = fmaf(sr, aip,  cr * aiq);
          A[i][p] = nip; A[p][i] = nip;
          A[i][q] = niq; A[q][i] = niq;
        }
        #pragma unroll
        for (int i = 0; i < 8; ++i) {
          const float vip = V[i][p], viq = V[i][q];
          V[i][p] = fmaf(cr, vip, -sr * viq);
          V[i][q] = fmaf(sr, vip,  cr * viq);
        }
      }
    }
  }

  // clamp eigenvalues, rebuild PD = V diag(lam) V^T, store
  float lam[8];
  #pragma unroll
  for (int k = 0; k < 8; ++k) lam[k] = fmaxf(A[k][k], 0.001f);

  float* pd = out + (size_t)LOC_ELEMS + gpos * 64;
  #pragma unroll
  for (int i = 0; i < 8; ++i) {
    float vl[8];
    #pragma unroll
    for (int k = 0; k < 8; ++k) vl[k] = V[i][k] * lam[k];
    float row[8];
    #pragma unroll
    for (int j = 0; j < 8; ++j) {
      float s0 = 0.0f;
      #pragma unroll
      for (int k = 0; k < 8; ++k) s0 = fmaf(vl[k], V[j][k], s0);
      row[j] = s0;
    }
    *(float4*)(pd + i * 8)     = make_float4(row[0], row[1], row[2], row[3]);
    *(float4*)(pd + i * 8 + 4) = make_float4(row[4], row[5], row[6], row[7]);
  }
}

extern "C" void kernel_launch(void* const* d_in, const int* in_sizes, int n_in,
                              void* d_out, int out_size, void* d_ws, size_t ws_size,
                              hipStream_t stream) {
  (void)in_sizes; (void)n_in; (void)out_size; (void)d_ws; (void)ws_size;
  const float* x   = (const float*)d_in[0];
  const float* w_n = (const float*)d_in[1];
  const float* b_n = (const float*)d_in[2];
  const float* w_p = (const float*)d_in[3];
  const float* b_p = (const float*)d_in[4];
  float* out = (float*)d_out;
  fused_gauss_kernel<<<dim3(512), dim3(256), 0, stream>>>(x, w_n, b_n, w_p, b_p, out);
}